// MaskBevEncoder_40072044871932
// MI455X (gfx1250) — compile-verified
//
#include <hip/hip_runtime.h>
#include <hip/hip_bf16.h>

typedef __attribute__((ext_vector_type(2))) float v2f;
typedef __attribute__((ext_vector_type(8))) float v8f;

#define BB     8
#define VV     200000
#define NN     32
#define CC     64
#define NXX    432
#define NYY    496
#define VXF    0.16f
#define VYF    0.16f
#define XOFF   0.08f          /* VX/2 + X_MIN */
#define YOFF   (-39.60f)      /* VY/2 + Y_MIN */
#define BN_EPS 1e-3f
#define LN_EPS 1e-3f
#define CELLS  (BB * NYY * NXX)            /* 1,714,176 */
#define CNT_F  (64.0f * 496.0f * 432.0f)   /* elements per LN group */

/* workspace layout (bytes) */
#define WS_XMAX_OFF 0
#define WS_IMAP_OFF 51200000               /* V*64*4 */
#define WS_STAT_OFF 58056704               /* + CELLS*4, 16B aligned */
/* stats: [0..7]=sum  [8..15]=sumsq  [16..23]=mu  [24..31]=inv_std */

__global__ __launch_bounds__(256) void init_kernel(int* __restrict__ imap,
                                                   float* __restrict__ stats) {
  unsigned i = blockIdx.x * 256u + threadIdx.x;
  if (i < CELLS) imap[i] = -1;
  if (i < 32) stats[i] = 0.0f;
}

__global__ __launch_bounds__(256) void pillar_kernel(
    const float* __restrict__ voxels, const int* __restrict__ nump,
    const int* __restrict__ coors,  const float* __restrict__ w,
    const float* __restrict__ gam,  const float* __restrict__ bet,
    const float* __restrict__ bmu,  const float* __restrict__ bvar,
    float* __restrict__ xmax, int* __restrict__ imap, float* __restrict__ stats)
{
  __shared__ __align__(16) float lds[8 * NN * 12];
  const int lane = threadIdx.x & 31;
  const int wave = threadIdx.x >> 5;
  const int v = blockIdx.x * 8 + wave;          /* grid is exactly V/8 */
  float* fl = &lds[wave * NN * 12];

  /* ---- per-point augmented features (lane = point) ---- */
  const float4 p4 = ((const float4*)voxels)[v * NN + lane];
  const int   np  = nump[v];
  const float mk  = (lane < np) ? 1.0f : 0.0f;
  const float px = p4.x * mk, py = p4.y * mk, pz = p4.z * mk, pw = p4.w * mk;

  float sx = px, sy = py, sz = pz;
  #pragma unroll
  for (int m = 16; m >= 1; m >>= 1) {
    sx += __shfl_xor(sx, m, 32);
    sy += __shfl_xor(sy, m, 32);
    sz += __shfl_xor(sz, m, 32);
  }
  const float inp = 1.0f / (float)np;
  const float mx = sx * inp, my = sy * inp, mz = sz * inp;

  const int cb  = coors[v * 4 + 0];
  const int ciy = coors[v * 4 + 2];
  const int cix = coors[v * 4 + 3];
  const float cx = (float)cix * VXF + XOFF;
  const float cy = (float)ciy * VYF + YOFF;

  float f[12];
  f[0] = px; f[1] = py; f[2] = pz; f[3] = pw;
  f[4] = (px - mx) * mk; f[5] = (py - my) * mk; f[6] = (pz - mz) * mk;
  f[7] = (px - cx) * mk; f[8] = (py - cy) * mk;
  f[9] = 0.0f; f[10] = 0.0f; f[11] = 0.0f;

  /* stage A-matrix rows (point-major, K padded to 12) into LDS */
  #pragma unroll
  for (int i = 0; i < 3; ++i)
    ((float4*)fl)[lane * 3 + i] =
        make_float4(f[4*i+0], f[4*i+1], f[4*i+2], f[4*i+3]);
  __syncthreads();

  /* ---- B fragments: w is [9 x 64], pad K to 12 ---- */
  const int nl = lane & 15;
  const int kh = (lane >> 4) << 1;              /* 0 or 2 */
  v2f bf[4][3];
  #pragma unroll
  for (int n = 0; n < 4; ++n)
    #pragma unroll
    for (int kk = 0; kk < 3; ++kk) {
      const int K0 = 4 * kk + kh;
      const int Nc = n * 16 + nl;
      const float b0 = (K0     < 9) ? w[K0 * 64 + Nc]       : 0.0f;
      const float b1 = (K0 + 1 < 9) ? w[(K0 + 1) * 64 + Nc] : 0.0f;
      v2f t; t.x = b0; t.y = b1; bf[n][kk] = t;
    }

  /* BN folded to scale/shift, per lane channel = n*16 + (lane&15) */
  float bsc[4], bsh[4];
  #pragma unroll
  for (int n = 0; n < 4; ++n) {
    const int c = n * 16 + nl;
    const float sc = gam[c] * rsqrtf(bvar[c] + BN_EPS);
    bsc[n] = sc;
    bsh[n] = bet[c] - bmu[c] * sc;
  }

  float cmax[4] = {-INFINITY, -INFINITY, -INFINITY, -INFINITY};

  #pragma unroll
  for (int t = 0; t < 2; ++t) {
    const int row = t * 16 + nl;
    v2f af[3];
    #pragma unroll
    for (int kk = 0; kk < 3; ++kk)
      af[kk] = *(const v2f*)&fl[row * 12 + 4 * kk + kh];
    #pragma unroll
    for (int n = 0; n < 4; ++n) {
      v8f acc = {0.f, 0.f, 0.f, 0.f, 0.f, 0.f, 0.f, 0.f};
      #pragma unroll
      for (int kk = 0; kk < 3; ++kk)
        acc = __builtin_amdgcn_wmma_f32_16x16x4_f32(
            false, af[kk], false, bf[n][kk], (short)0, acc, false, false);
      #pragma unroll
      for (int r = 0; r < 8; ++r) {
        const float xv = fmaxf(acc[r] * bsc[n] + bsh[n], 0.0f);
        cmax[n] = fmaxf(cmax[n], xv);
      }
    }
  }

  /* merge the two row-halves (lanes l and l+16 hold same channels) */
  #pragma unroll
  for (int n = 0; n < 4; ++n)
    cmax[n] = fmaxf(cmax[n], __shfl_xor(cmax[n], 16, 32));

  /* LN statistics contribution of this pillar */
  float s = 0.0f, s2 = 0.0f;
  #pragma unroll
  for (int n = 0; n < 4; ++n) { s += cmax[n]; s2 += cmax[n] * cmax[n]; }
  #pragma unroll
  for (int m = 8; m >= 1; m >>= 1) {
    s  += __shfl_xor(s,  m, 32);
    s2 += __shfl_xor(s2, m, 32);
  }

  if (lane < 16) {
    #pragma unroll
    for (int n = 0; n < 4; ++n)
      xmax[v * CC + n * 16 + nl] = cmax[n];
  }
  if (lane == 0) {
    atomicAdd(&stats[cb], s);
    atomicAdd(&stats[8 + cb], s2);
    imap[cb * (NYY * NXX) + ciy * NXX + cix] = v;
  }
}

__global__ void stats_kernel(float* __restrict__ stats) {
  const int b = threadIdx.x;
  if (b < BB) {
    const float mu  = stats[b] / CNT_F;
    const float var = stats[8 + b] / CNT_F - mu * mu;
    stats[16 + b] = mu;
    stats[24 + b] = rsqrtf(var + LN_EPS);
  }
}

__global__ __launch_bounds__(256) void out_kernel(
    const float* __restrict__ lnw, const float* __restrict__ lnb,
    const float* __restrict__ xmax, const int* __restrict__ imap,
    const float* __restrict__ stats, float* __restrict__ out)
{
  const unsigned X4 = NXX / 4;                      /* 108 */
  const unsigned total = (unsigned)BB * CC * NYY * X4;
  unsigned gid = blockIdx.x * 256u + threadIdx.x;
  if (gid >= total) return;
  const unsigned x4 = gid % X4; unsigned r = gid / X4;
  const unsigned y  = r % NYY;  r /= NYY;
  const unsigned c  = r % CC;
  const unsigned b  = r / CC;

  const int4   pm = ((const int4*)imap)[(b * NYY + y) * X4 + x4];
  const float4 wv = ((const float4*)lnw)[(c * NYY + y) * X4 + x4];
  const float4 bv = ((const float4*)lnb)[(c * NYY + y) * X4 + x4];
  const float mu = stats[16 + b], inv = stats[24 + b];

  const float v0 = (pm.x >= 0) ? xmax[pm.x * CC + c] : 0.0f;
  const float v1 = (pm.y >= 0) ? xmax[pm.y * CC + c] : 0.0f;
  const float v2 = (pm.z >= 0) ? xmax[pm.z * CC + c] : 0.0f;
  const float v3 = (pm.w >= 0) ? xmax[pm.w * CC + c] : 0.0f;

  float4 o;
  o.x = (v0 - mu) * inv * wv.x + bv.x;
  o.y = (v1 - mu) * inv * wv.y + bv.y;
  o.z = (v2 - mu) * inv * wv.z + bv.z;
  o.w = (v3 - mu) * inv * wv.w + bv.w;
  ((float4*)out)[gid] = o;
}

extern "C" void kernel_launch(void* const* d_in, const int* in_sizes, int n_in,
                              void* d_out, int out_size, void* d_ws, size_t ws_size,
                              hipStream_t stream) {
  const float* voxels = (const float*)d_in[0];
  const int*   nump   = (const int*)d_in[1];
  const int*   coors  = (const int*)d_in[2];
  const float* w      = (const float*)d_in[3];
  const float* gam    = (const float*)d_in[4];
  const float* bet    = (const float*)d_in[5];
  const float* bmu    = (const float*)d_in[6];
  const float* bvar   = (const float*)d_in[7];
  const float* lnw    = (const float*)d_in[8];
  const float* lnb    = (const float*)d_in[9];
  float* out = (float*)d_out;

  char* ws = (char*)d_ws;
  float* xmax  = (float*)(ws + WS_XMAX_OFF);
  int*   imap  = (int*)(ws + WS_IMAP_OFF);
  float* stats = (float*)(ws + WS_STAT_OFF);

  init_kernel<<<(CELLS + 255) / 256, 256, 0, stream>>>(imap, stats);
  pillar_kernel<<<VV / 8, 256, 0, stream>>>(voxels, nump, coors, w, gam, bet,
                                            bmu, bvar, xmax, imap, stats);
  stats_kernel<<<1, 32, 0, stream>>>(stats);
  const unsigned groups = (unsigned)BB * CC * NYY * (NXX / 4);
  out_kernel<<<(groups + 255) / 256, 256, 0, stream>>>(lnw, lnb, xmax, imap,
                                                       stats, out);
}